// GINLayer_17635135718112
// MI455X (gfx1250) — compile-verified
//
#include <hip/hip_runtime.h>

typedef __attribute__((ext_vector_type(2))) float v2f;
typedef __attribute__((ext_vector_type(8))) float v8f;

#define N_NODES 50000
#define N_EDGES 1600000
#define D 128
#define EPS 0.001f
#define N_TILES (N_NODES / 16)      // 3125 (exact)
#define WAVES_PER_BLOCK 4

// ---------------------------------------------------------------------------
// Kernel 1: zero the aggregation scratch buffer (re-run every launch so graph
// replay stays deterministic).
// ---------------------------------------------------------------------------
__global__ void gin_zero_agg(float* __restrict__ agg, int n) {
  int i = blockIdx.x * blockDim.x + threadIdx.x;
  if (i < n) agg[i] = 0.0f;
}

// ---------------------------------------------------------------------------
// Kernel 2: edge scatter-add.  Thread t handles edge t/32, feature chunk
// (t%32)*4 .. +3.  x and agg are L2-resident (25.6 MB each), so the gather
// and the f32 atomics stay on-chip.
// ---------------------------------------------------------------------------
__global__ void gin_edge_scatter(const float* __restrict__ x,
                                 const long long* __restrict__ src,
                                 const long long* __restrict__ dst,
                                 float* __restrict__ agg) {
  long long gid = (long long)blockIdx.x * blockDim.x + threadIdx.x;
  int e = (int)(gid >> 5);
  if (e >= N_EDGES) return;
  int c = ((int)gid & 31) * 4;

  int s = (int)src[e];
  int d = (int)dst[e];

  const float4 v = *(const float4*)(x + (long long)s * D + c);   // global_load_b128
  float* p = agg + (long long)d * D + c;
  atomicAdd(p + 0, v.x);   // global_atomic_add_f32 (no return)
  atomicAdd(p + 1, v.y);
  atomicAdd(p + 2, v.z);
  atomicAdd(p + 3, v.w);
}

// ---------------------------------------------------------------------------
// Kernel 3: fused  h = (1+eps)x + agg ;  h1 = relu(h@W1+b1) ; out = h1@W2+b2
// One wave32 owns a 16-row node tile.  WMMA f32 16x16x4 fragments:
//   A (16x4):  lane l<16 -> {h[l][k],   h[l][k+1]};  l>=16 -> {h[l-16][k+2], h[l-16][k+3]}
//   B (4x16):  lane l<16 -> {W[k][n0+l], W[k+1][n0+l]}; l>=16 -> rows k+2,k+3
//   C (16x16): VGPR r, lane l<16 -> (M=r,   N=n0+l); l>=16 -> (M=r+8, N=n0+l-16)
// ---------------------------------------------------------------------------
__global__ __launch_bounds__(WAVES_PER_BLOCK * 32)
void gin_mlp(const float* __restrict__ x, const float* __restrict__ agg,
             const float* __restrict__ W1, const float* __restrict__ b1,
             const float* __restrict__ W2, const float* __restrict__ b2,
             float* __restrict__ out) {
  __shared__ float lds_h [WAVES_PER_BLOCK][16 * D];
  __shared__ float lds_h1[WAVES_PER_BLOCK][16 * D];

  const int lane = threadIdx.x & 31;
  const int wave = threadIdx.x >> 5;
  const int tile = blockIdx.x * WAVES_PER_BLOCK + wave;   // wave-uniform guard
  if (tile >= N_TILES) return;                            // EXEC stays all-1s

  const int m0 = tile * 16;
  float* h  = lds_h[wave];
  float* h1 = lds_h1[wave];

  // ---- stage h = (1+eps)*x + agg into LDS (wave-private, DS in-order) ----
  for (int r = 0; r < 16; ++r) {
    const long long base = (long long)(m0 + r) * D + lane * 4;
    float4 xv = *(const float4*)(x + base);
    float4 av = *(const float4*)(agg + base);
    float4 hv;
    hv.x = (1.0f + EPS) * xv.x + av.x;
    hv.y = (1.0f + EPS) * xv.y + av.y;
    hv.z = (1.0f + EPS) * xv.z + av.z;
    hv.w = (1.0f + EPS) * xv.w + av.w;
    *(float4*)(h + r * D + lane * 4) = hv;
  }

  const int al    = lane & 15;          // column within N-tile / row within M-tile
  const int koff  = (lane >> 4) << 1;   // 0 for lanes 0-15, 2 for lanes 16-31
  const int rbase = (lane >> 4) << 3;   // 0 or 8 (C-matrix M offset)

  // ---- layer 1: h1 = relu(h @ W1 + b1) ----
  for (int nt = 0; nt < 8; ++nt) {
    const int n = nt * 16 + al;
    const float bv = b1[n];
    v8f acc = {bv, bv, bv, bv, bv, bv, bv, bv};
#pragma unroll 4
    for (int k = 0; k < D; k += 4) {
      v2f a, b;
      a.x = h[al * D + k + koff];       // ds_load_b64 (consecutive pair)
      a.y = h[al * D + k + koff + 1];
      b.x = W1[(k + koff) * D + n];
      b.y = W1[(k + koff + 1) * D + n];
      acc = __builtin_amdgcn_wmma_f32_16x16x4_f32(
          false, a, false, b, (short)0, acc, false, false);
    }
#pragma unroll
    for (int r = 0; r < 8; ++r) {
      float v = acc[r];
      h1[(rbase + r) * D + n] = v > 0.0f ? v : 0.0f;      // ReLU
    }
  }

  // ---- layer 2: out = h1 @ W2 + b2 ----
  for (int nt = 0; nt < 8; ++nt) {
    const int n = nt * 16 + al;
    const float bv = b2[n];
    v8f acc = {bv, bv, bv, bv, bv, bv, bv, bv};
#pragma unroll 4
    for (int k = 0; k < D; k += 4) {
      v2f a, b;
      a.x = h1[al * D + k + koff];
      a.y = h1[al * D + k + koff + 1];
      b.x = W2[(k + koff) * D + n];
      b.y = W2[(k + koff + 1) * D + n];
      acc = __builtin_amdgcn_wmma_f32_16x16x4_f32(
          false, a, false, b, (short)0, acc, false, false);
    }
#pragma unroll
    for (int r = 0; r < 8; ++r) {
      out[(long long)(m0 + rbase + r) * D + n] = acc[r];
    }
  }
}

// ---------------------------------------------------------------------------
extern "C" void kernel_launch(void* const* d_in, const int* in_sizes, int n_in,
                              void* d_out, int out_size, void* d_ws, size_t ws_size,
                              hipStream_t stream) {
  const float*     x  = (const float*)d_in[0];
  const long long* ei = (const long long*)d_in[1];   // int64 (2, N_EDGES)
  const float*     W1 = (const float*)d_in[2];
  const float*     b1 = (const float*)d_in[3];
  const float*     W2 = (const float*)d_in[4];
  const float*     b2 = (const float*)d_in[5];
  float*           out = (float*)d_out;

  const long long* src = ei;             // row 0
  const long long* dst = ei + N_EDGES;   // row 1

  float* agg = (float*)d_ws;             // N_NODES * D floats = 25.6 MB

  // 1) zero aggregation buffer
  {
    int n = N_NODES * D;
    int blocks = (n + 255) / 256;
    gin_zero_agg<<<blocks, 256, 0, stream>>>(agg, n);
  }
  // 2) scatter-add over edges (one thread per edge x 4-feature chunk)
  {
    long long threads = (long long)N_EDGES * 32;
    int blocks = (int)((threads + 255) / 256);
    gin_edge_scatter<<<blocks, 256, 0, stream>>>(x, src, dst, agg);
  }
  // 3) fused GIN MLP with WMMA f32 16x16x4
  {
    int blocks = (N_TILES + WAVES_PER_BLOCK - 1) / WAVES_PER_BLOCK;
    gin_mlp<<<blocks, WAVES_PER_BLOCK * 32, 0, stream>>>(x, agg, W1, b1, W2, b2, out);
  }
}